// MultiBoxLoss_53669911330898
// MI455X (gfx1250) — compile-verified
//
#include <hip/hip_runtime.h>

typedef __attribute__((ext_vector_type(2))) float v2f;
typedef __attribute__((ext_vector_type(8))) float v8f;

#define B_ 32
#define P_ 24564
#define C_ 81
#define M_ 16

__device__ __forceinline__ float smooth_l1(float x) {
    float ax = fabsf(x);
    return (ax < 1.f) ? 0.5f * x * x : ax - 0.5f;
}

// ---------------------------------------------------------------- init
__global__ void init_kernel(int* __restrict__ forced,
                            unsigned long long* __restrict__ packed,
                            int* __restrict__ n_pos,
                            float* __restrict__ parts,
                            long BP, int nbm, int nparts) {
    long i = (long)blockIdx.x * blockDim.x + threadIdx.x;
    if (i < BP) forced[i] = -1;
    if (i < nbm) packed[i] = 0ull;
    if (i < B_) n_pos[i] = 0;
    if (i < nparts) parts[i] = 0.f;
}

// ---------------------------------------------------------------- match
__global__ void match_kernel(const float* __restrict__ boxes,
                             const float* __restrict__ priors,
                             float* __restrict__ best_iou,
                             int* __restrict__ best_obj,
                             unsigned long long* __restrict__ packed) {
    __shared__ unsigned long long s_pack[M_];
    int b = blockIdx.y;
    int p = blockIdx.x * blockDim.x + threadIdx.x;
    if (threadIdx.x < M_) s_pack[threadIdx.x] = 0ull;
    __syncthreads();
    if (p < P_) {
        float4 pr = ((const float4*)priors)[p];
        float px1 = pr.x - pr.z * 0.5f, py1 = pr.y - pr.w * 0.5f;
        float px2 = pr.x + pr.z * 0.5f, py2 = pr.y + pr.w * 0.5f;
        float parea = pr.z * pr.w;
        float biou = -1.f; int bm = 0;
        #pragma unroll
        for (int m = 0; m < M_; ++m) {
            float4 bx = ((const float4*)boxes)[b * M_ + m];
            float lx = fmaxf(bx.x, px1), ly = fmaxf(bx.y, py1);
            float rx = fminf(bx.z, px2), ry = fminf(bx.w, py2);
            float iw = fmaxf(rx - lx, 0.f), ih = fmaxf(ry - ly, 0.f);
            float inter = iw * ih;
            float barea = (bx.z - bx.x) * (bx.w - bx.y);
            float iou = inter / (barea + parea - inter);
            if (iou > biou) { biou = iou; bm = m; }   // strict > : first index wins
            unsigned long long pk =
                ((unsigned long long)__float_as_uint(iou) << 32) |
                (unsigned long long)(0xFFFFFFFFu - (unsigned)p); // ties -> smallest p
            atomicMax(&s_pack[m], pk);
        }
        long idx = (long)b * P_ + p;
        best_iou[idx] = biou;
        best_obj[idx] = bm;
    }
    __syncthreads();
    if (threadIdx.x < M_)
        atomicMax(&packed[b * M_ + threadIdx.x], s_pack[threadIdx.x]);
}

// ---------------------------------------------------------------- scatter forced matches
__global__ void scatter_kernel(const unsigned long long* __restrict__ packed,
                               int* __restrict__ forced) {
    int i = blockIdx.x * blockDim.x + threadIdx.x;
    if (i < B_ * M_) {
        int b = i >> 4, m = i & 15;
        unsigned p = 0xFFFFFFFFu - (unsigned)(packed[i] & 0xFFFFFFFFull);
        forced[(long)b * P_ + p] = m;
    }
}

// ---------------------------------------------------------------- main streaming kernel
__global__ void main_kernel(const float* __restrict__ locs,
                            const float* __restrict__ scores,
                            const float* __restrict__ boxes,
                            const int* __restrict__ labels,
                            const float* __restrict__ priors,
                            const float* __restrict__ best_iou,
                            const int* __restrict__ best_obj,
                            const int* __restrict__ forced,
                            float* __restrict__ conf_neg,
                            int* __restrict__ n_pos,
                            float* __restrict__ confpos_part,
                            float* __restrict__ loc_part) {
    long idx = (long)blockIdx.x * blockDim.x + threadIdx.x;
    const long BP = (long)B_ * P_;
    float cp = 0.f, ll = 0.f;
    if (idx < BP) {
        int b = (int)(idx / P_);
        int p = (int)(idx - (long)b * P_);
        int obj = forced[idx];
        float iou;
        if (obj >= 0) { iou = 1.f; }
        else          { obj = best_obj[idx]; iou = best_iou[idx]; }
        int lab = labels[b * M_ + obj];
        int cls = (iou < 0.5f) ? 0 : lab;
        bool pos = (cls != 0);

        const float* srow = scores + idx * C_;
        __builtin_prefetch(srow + (size_t)C_ * 256, 0, 1);  // gfx1250 global_prefetch_b8

        // single-pass online logsumexp over 81 classes
        float mx = srow[0];
        float s = 1.f;
        for (int j = 1; j < C_; ++j) {
            float x = srow[j];
            float nm = fmaxf(mx, x);
            s = s * expf(mx - nm) + expf(x - nm);
            mx = nm;
        }
        float ce = logf(s) + mx - srow[cls];

        conf_neg[idx] = pos ? 0.f : ce;
        cp = pos ? ce : 0.f;

        if (pos) {
            atomicAdd(&n_pos[b], 1);  // integer atomic: deterministic
            float4 bx = ((const float4*)boxes)[b * M_ + obj];
            float4 pr = ((const float4*)priors)[p];
            float cx = (bx.x + bx.z) * 0.5f, cy = (bx.y + bx.w) * 0.5f;
            float w = bx.z - bx.x, h = bx.w - bx.y;
            float g0 = (cx - pr.x) / (pr.z * 0.1f);
            float g1 = (cy - pr.y) / (pr.w * 0.1f);
            float g2 = logf(w / pr.z) * 5.f;
            float g3 = logf(h / pr.w) * 5.f;
            float4 pl = ((const float4*)locs)[idx];
            ll = smooth_l1(pl.x - g0) + smooth_l1(pl.y - g1) +
                 smooth_l1(pl.z - g2) + smooth_l1(pl.w - g3);
        }
    }
    // fixed-order block reduction (deterministic)
    __shared__ float sA[8], sB[8];
    #pragma unroll
    for (int off = 16; off > 0; off >>= 1) {
        cp += __shfl_down(cp, off, 32);
        ll += __shfl_down(ll, off, 32);
    }
    int wid = threadIdx.x >> 5, lane = threadIdx.x & 31;
    if (lane == 0) { sA[wid] = cp; sB[wid] = ll; }
    __syncthreads();
    if (threadIdx.x == 0) {
        float a = 0.f, c = 0.f;
        #pragma unroll
        for (int i = 0; i < 8; ++i) { a += sA[i]; c += sB[i]; }
        confpos_part[blockIdx.x] = a;
        loc_part[blockIdx.x] = c;
    }
}

// ---------------------------------------------------------------- exact per-image top-k sum
__global__ __launch_bounds__(1024)
void hardneg_kernel(const float* __restrict__ conf_neg,
                    const int* __restrict__ n_pos,
                    float* __restrict__ hard_sum) {
    const int ITEMS = (P_ + 1023) / 1024;  // 24
    int b = blockIdx.x;
    int tid = threadIdx.x;
    long base = (long)b * P_;
    float v[ITEMS];
    unsigned key[ITEMS];
    #pragma unroll
    for (int i = 0; i < ITEMS; ++i) {
        int p = tid + i * 1024;
        float x = (p < P_) ? conf_neg[base + p] : 0.f;  // pad zeros are harmless (CE>=0)
        v[i] = x;
        key[i] = __float_as_uint(x);
    }
    int k = n_pos[b] * 3;
    if (k > P_) k = P_;
    __shared__ int s_i[32];
    __shared__ float s_f[32];
    __shared__ int s_bc;
    if (k <= 0) { if (tid == 0) hard_sum[b] = 0.f; return; }  // uniform per block

    // binary search in uint space for k-th largest: smallest t with count(>t) < k
    unsigned lo = 0u, hi = 0xFFFFFFFFu;
    while (lo < hi) {
        unsigned mid = lo + ((hi - lo) >> 1);
        int c = 0;
        #pragma unroll
        for (int i = 0; i < ITEMS; ++i) c += (key[i] > mid) ? 1 : 0;
        #pragma unroll
        for (int off = 16; off > 0; off >>= 1) c += __shfl_down(c, off, 32);
        if ((tid & 31) == 0) s_i[tid >> 5] = c;
        __syncthreads();
        if (tid == 0) {
            int t = 0;
            #pragma unroll
            for (int i = 0; i < 32; ++i) t += s_i[i];
            s_bc = t;
        }
        __syncthreads();
        c = s_bc;
        if (c >= k) lo = mid + 1; else hi = mid;
        __syncthreads();  // protect s_i reuse next iteration
    }
    unsigned t = lo;
    float tf = __uint_as_float(t);
    int c = 0; float s = 0.f;
    #pragma unroll
    for (int i = 0; i < ITEMS; ++i) {
        if (key[i] > t) { c += 1; s += v[i]; }
    }
    #pragma unroll
    for (int off = 16; off > 0; off >>= 1) {
        c += __shfl_down(c, off, 32);
        s += __shfl_down(s, off, 32);
    }
    if ((tid & 31) == 0) { s_i[tid >> 5] = c; s_f[tid >> 5] = s; }
    __syncthreads();
    if (tid == 0) {
        int ct = 0; float st = 0.f;
        #pragma unroll
        for (int i = 0; i < 32; ++i) { ct += s_i[i]; st += s_f[i]; }
        hard_sum[b] = st + tf * (float)(k - ct);  // exact with duplicates
    }
}

// ---------------------------------------------------------------- final: WMMA row-sum reduction
// D = A x ones(4x16) + C : every column of D gets the 16x4 row-sums -> full f32
// reduction with hardware-fixed (deterministic) summation order.
__global__ void final_kernel(const float* __restrict__ confpos_part,
                             const float* __restrict__ loc_part,
                             const float* __restrict__ hard_sum,
                             const int* __restrict__ n_pos,
                             float* __restrict__ out, int nchunks) {
    int lane = threadIdx.x;          // single wave32, EXEC all ones
    int m = lane & 15;               // A 16x4 layout: lanes 0-15 K=0,1; lanes 16-31 K=2,3
    int kb = (lane >> 4) << 1;
    v8f accC = {0.f, 0.f, 0.f, 0.f, 0.f, 0.f, 0.f, 0.f};
    v8f accL = {0.f, 0.f, 0.f, 0.f, 0.f, 0.f, 0.f, 0.f};
    v2f ones; ones.x = 1.f; ones.y = 1.f;
    for (int c = 0; c < nchunks; ++c) {
        int base = c * 64 + m * 4 + kb;
        v2f a; a.x = confpos_part[base]; a.y = confpos_part[base + 1];
        accC = __builtin_amdgcn_wmma_f32_16x16x4_f32(false, a, false, ones,
                                                     (short)0, accC, false, false);
        v2f l; l.x = loc_part[base]; l.y = loc_part[base + 1];
        accL = __builtin_amdgcn_wmma_f32_16x16x4_f32(false, l, false, ones,
                                                     (short)0, accL, false, false);
    }
    // column N=0 lives in lane 0 (M=0..7) and lane 16 (M=8..15)
    float sC = 0.f, sL = 0.f;
    #pragma unroll
    for (int i = 0; i < 8; ++i) { sC += accC[i]; sL += accL[i]; }
    sC = __shfl(sC, 0, 32) + __shfl(sC, 16, 32);
    sL = __shfl(sL, 0, 32) + __shfl(sL, 16, 32);
    float hard = 0.f; int npt = 0;
    #pragma unroll
    for (int b = 0; b < B_; ++b) { hard += hard_sum[b]; npt += n_pos[b]; }
    if (lane == 0) {
        float np = (float)npt;
        out[0] = (sC + hard) / np + sL / (4.f * np);
    }
}

// ---------------------------------------------------------------- launch
extern "C" void kernel_launch(void* const* d_in, const int* in_sizes, int n_in,
                              void* d_out, int out_size, void* d_ws, size_t ws_size,
                              hipStream_t stream) {
    const long BP = (long)B_ * P_;                 // 786048
    const int NBLK = (int)((BP + 255) / 256);      // 3071
    const int NBLK_PAD = ((NBLK + 63) / 64) * 64;  // 3072

    const float* locs   = (const float*)d_in[0];
    const float* scores = (const float*)d_in[1];
    const float* boxes  = (const float*)d_in[2];
    const int*   labels = (const int*)d_in[3];
    const float* priors = (const float*)d_in[4];
    float* out = (float*)d_out;

    char* w = (char*)d_ws;
    float* conf_neg = (float*)w;                          w += BP * sizeof(float);
    float* best_iou = (float*)w;                          w += BP * sizeof(float);
    int*   best_obj = (int*)w;                            w += BP * sizeof(int);
    int*   forced   = (int*)w;                            w += BP * sizeof(int);
    unsigned long long* packed = (unsigned long long*)w;  w += B_ * M_ * sizeof(unsigned long long);
    int*   n_pos = (int*)w;                               w += B_ * sizeof(int);
    float* confpos_part = (float*)w;                      w += NBLK_PAD * sizeof(float);
    float* loc_part = (float*)w;                          w += NBLK_PAD * sizeof(float);
    float* hard_sum = (float*)w;                          w += B_ * sizeof(float);

    init_kernel<<<(unsigned)((BP + 255) / 256), 256, 0, stream>>>(
        forced, packed, n_pos, confpos_part, BP, B_ * M_, 2 * NBLK_PAD);

    dim3 g1((P_ + 255) / 256, B_);
    match_kernel<<<g1, 256, 0, stream>>>(boxes, priors, best_iou, best_obj, packed);

    scatter_kernel<<<2, 256, 0, stream>>>(packed, forced);

    main_kernel<<<NBLK, 256, 0, stream>>>(locs, scores, boxes, labels, priors,
                                          best_iou, best_obj, forced,
                                          conf_neg, n_pos, confpos_part, loc_part);

    hardneg_kernel<<<B_, 1024, 0, stream>>>(conf_neg, n_pos, hard_sum);

    final_kernel<<<1, 32, 0, stream>>>(confpos_part, loc_part, hard_sum, n_pos,
                                       out, NBLK_PAD / 64);
}